// SATCN_31044023616121
// MI455X (gfx1250) — compile-verified
//
#include <hip/hip_runtime.h>
#include <math.h>

#define BB 4
#define TT 12
#define NN 2000
#define KK 8
#define CC 32
#define PP (BB*TT*NN)   // 96000 positions
#define WPB 4           // waves per block in layer1

typedef __attribute__((ext_vector_type(2))) float v2f;
typedef __attribute__((ext_vector_type(8))) float v8f;

__device__ __forceinline__ v8f wmma4(v2f a, v2f b, v8f c) {
  // D(16x16,f32) += A(16x4,f32) * B(4x16,f32)
  return __builtin_amdgcn_wmma_f32_16x16x4_f32(false, a, false, b, (short)0, c, false, false);
}

// ---------------- Kernel 1: per-row top-K (sparsify) ----------------
// One wave per row; 8 passes of wave-wide argmax with exclusion list.
__global__ void __launch_bounds__(256)
topk_kernel(const float* __restrict__ adj, int* __restrict__ topi,
            float* __restrict__ topv, float* __restrict__ degw,
            float* __restrict__ srow) {
  const int lane = threadIdx.x & 31;
  const int row  = blockIdx.x * 8 + (threadIdx.x >> 5);
  const float* arow = adj + (size_t)row * NN;
  int   sel[KK];
  float val[KK];
  #pragma unroll
  for (int k = 0; k < KK; ++k) sel[k] = -1;
  for (int k = 0; k < KK; ++k) {
    float bestv = -1.0f; int besti = 0x7fffffff;
    for (int m = lane; m < NN; m += 32) {
      bool skip = false;
      #pragma unroll
      for (int j = 0; j < KK; ++j) skip = skip || (sel[j] == m);
      float v = skip ? -1.0f : arow[m];
      if (v > bestv) { bestv = v; besti = m; }
    }
    #pragma unroll
    for (int off = 16; off > 0; off >>= 1) {
      float ov = __shfl_xor(bestv, off, 32);
      int   oi = __shfl_xor(besti, off, 32);
      if (ov > bestv || (ov == bestv && oi < besti)) { bestv = ov; besti = oi; }
    }
    sel[k] = besti; val[k] = bestv;
  }
  if (lane == 0) {
    float dw = 0.f; int cnt = 0;
    #pragma unroll
    for (int k = 0; k < KK; ++k) {
      topi[row*KK + k] = sel[k];
      topv[row*KK + k] = val[k];
      dw  += val[k];
      cnt += (val[k] > 0.f) ? 1 : 0;
    }
    degw[row] = dw;
    if (srow) srow[row] = log1pf((float)cnt);   // scaler s = log1p(deg-count)
  }
}

// ---------------- Kernel 2: stower0 (masking) + tcn0, fused ----------------
// One thread per (b,t,n); writes h1 position-major [p][c] so stage-2 neighbor
// gathers are contiguous 128B vectors.
__global__ void __launch_bounds__(256)
layer0_kernel(const float* __restrict__ X, const int* __restrict__ topi,
              const float* __restrict__ topv, const float* __restrict__ degw,
              const float* __restrict__ Theta0, const float* __restrict__ bias0,
              const float* __restrict__ Wt0, const float* __restrict__ bt0,
              float* __restrict__ h1) {
  __shared__ float sW[CC*CC];
  __shared__ float sTh[4*CC];
  __shared__ float sB0[CC];
  __shared__ float sBt[CC];
  for (int i = threadIdx.x; i < CC*CC; i += 256) sW[i] = Wt0[i];
  if (threadIdx.x < 4*CC) sTh[threadIdx.x] = Theta0[threadIdx.x];
  if (threadIdx.x < CC) { sB0[threadIdx.x] = bias0[threadIdx.x]; sBt[threadIdx.x] = bt0[threadIdx.x]; }
  __syncthreads();

  const int p    = blockIdx.x * 256 + threadIdx.x;   // p = (b*T+t)*N + n
  const int nrow = p % NN;
  const int bt   = p / NN;
  const float* xbt = X + (size_t)bt * NN;

  float ws = 0.f, wss = 0.f, mx = -INFINITY, mn = INFINITY;
  #pragma unroll
  for (int k = 0; k < KK; ++k) {
    int   m = topi[nrow*KK + k];
    float w = topv[nrow*KK + k];
    float xv = xbt[m];
    ws  += w * xv;
    wss += w * xv * xv;
    if (w > 0.f) { mx = fmaxf(mx, xv); mn = fminf(mn, xv); }
  }
  float dw   = degw[nrow];
  float mean = ws / dw;
  float sd   = sqrtf(fmaxf(wss / dw - mean*mean, 0.f) + 1e-5f);
  float m4[4] = {mean, mx, mn, sd};       // concat order: mean, max, min, std
  #pragma unroll
  for (int i = 0; i < 4; ++i) {           // masking: nan->6, inf->0, clamp 6
    float v = m4[i];
    v = isnan(v) ? 6.0f : v;
    v = isinf(v) ? 0.0f : v;
    m4[i] = fminf(v, 6.0f);
  }
  float v[CC];
  #pragma unroll
  for (int c = 0; c < CC; ++c) {
    float a = sB0[c];
    #pragma unroll
    for (int i = 0; i < 4; ++i) a = fmaf(m4[i], sTh[i*CC + c], a);
    v[c] = a;
  }
  float* out = h1 + (size_t)p * CC;
  #pragma unroll 4
  for (int o = 0; o < CC; ++o) {          // tcn0: tanh(W v + b + v)
    float a = sBt[o] + v[o];
    #pragma unroll
    for (int c = 0; c < CC; ++c) a = fmaf(sW[o*CC + c], v[c], a);
    out[o] = tanhf(a);
  }
}

// ---------------- Kernel 3: stower1 + Theta1 GEMM + tcn1 + Wout, fused ------
// One wave per 16-position tile. GEMMs run on V_WMMA_F32_16X16X4_F32.
__global__ void __launch_bounds__(128)
layer1_kernel(const float* __restrict__ h1, const int* __restrict__ topi,
              const float* __restrict__ topv, const float* __restrict__ degw,
              const float* __restrict__ srow,
              const float* __restrict__ Theta1, const float* __restrict__ bias1,
              const float* __restrict__ Wt1, const float* __restrict__ bt1,
              const float* __restrict__ Wout, const float* __restrict__ bout,
              float* __restrict__ y) {
  __shared__ float mlds[WPB][16*132];   // 16 pos x 128 agg features (pad 132: conflict-free)
  __shared__ float hlds[WPB][16*33];    // 16 pos x 32 ch, relu(stower1) output
  __shared__ float sb[WPB][16];         // per-position scaler s

  const int wv    = threadIdx.x >> 5;
  const int lane  = threadIdx.x & 31;
  const int tile  = blockIdx.x * WPB + wv;
  const int tbase = tile * 16;

  // ---- aggregation: 2 lanes per position, 16 features each, K=8 gather ----
  const int pos  = lane >> 1;
  const int f0   = (lane & 1) * 16;
  const int P0   = tbase + pos;
  const int nrow = P0 % NN;
  const int bt   = P0 / NN;

  float aws[16], awss[16], amx[16], amn[16];
  #pragma unroll
  for (int f = 0; f < 16; ++f) { aws[f]=0.f; awss[f]=0.f; amx[f]=-INFINITY; amn[f]=INFINITY; }
  #pragma unroll
  for (int k = 0; k < KK; ++k) {
    int   m = topi[nrow*KK + k];
    float w = topv[nrow*KK + k];
    const float4* src = (const float4*)(h1 + ((size_t)bt*NN + m)*CC + f0);
    bool valid = w > 0.f;
    #pragma unroll
    for (int q = 0; q < 4; ++q) {
      float4 d = src[q];
      float e[4] = {d.x, d.y, d.z, d.w};
      #pragma unroll
      for (int r = 0; r < 4; ++r) {
        int f = q*4 + r;
        aws[f]  += w * e[r];
        awss[f] += w * e[r] * e[r];
        if (valid) { amx[f] = fmaxf(amx[f], e[r]); amn[f] = fminf(amn[f], e[r]); }
      }
    }
  }
  {
    float dw = degw[nrow];
    float* mrow_w = &mlds[wv][pos*132];
    #pragma unroll
    for (int f = 0; f < 16; ++f) {
      float mean = aws[f]/dw;
      float sd = sqrtf(fmaxf(awss[f]/dw - mean*mean, 0.f) + 1e-5f);
      mrow_w[0*CC + f0 + f] = mean;
      mrow_w[1*CC + f0 + f] = amx[f];
      mrow_w[2*CC + f0 + f] = amn[f];
      mrow_w[3*CC + f0 + f] = sd;
    }
    if (f0 == 0) sb[wv][pos] = srow[nrow];
  }
  __syncthreads();

  // ---- GEMM1: D(16x32) = A(16x384) * Theta1(384x32); scalers folded into A ----
  const int lh  = lane >> 4;    // 0: K pair {0,1}, C rows 0-7 ; 1: K pair {2,3}, C rows 8-15
  const int l15 = lane & 15;
  const float* mrow = &mlds[wv][l15*132];
  const float s1 = sb[wv][l15];
  const float svec[3] = {1.f, s1, 1.f/s1};

  v8f acc0 = {}; v8f acc1 = {};
  #pragma unroll
  for (int g = 0; g < 3; ++g) {                 // groups: m, m*s, m/s
    const float sc = svec[g];
    #pragma unroll 4
    for (int kb = 0; kb < 32; ++kb) {
      const int kf = kb*4 + lh*2;               // feature within 128-block
      v2f a; a.x = mrow[kf]*sc; a.y = mrow[kf+1]*sc;
      const int kg = g*128 + kf;                // row index into Theta1 (384x32)
      v2f b0, b1;
      b0.x = Theta1[kg*CC + l15];       b0.y = Theta1[(kg+1)*CC + l15];
      b1.x = Theta1[kg*CC + 16 + l15];  b1.y = Theta1[(kg+1)*CC + 16 + l15];
      acc0 = wmma4(a, b0, acc0);
      acc1 = wmma4(a, b1, acc1);
    }
  }

  // bias + relu -> LDS tile (pos x 32)
  const float bb0 = bias1[l15], bb1 = bias1[16 + l15];
  #pragma unroll
  for (int vv = 0; vv < 8; ++vv) {
    int row = vv + lh*8;
    hlds[wv][row*33 + l15]      = fmaxf(acc0[vv] + bb0, 0.f);
    hlds[wv][row*33 + 16 + l15] = fmaxf(acc1[vv] + bb1, 0.f);
  }
  __syncthreads();

  // ---- GEMM2 (tcn1): T(16x32) = H(16x32) * Wt1^T(32x32) ----
  v8f t0 = {}; v8f t1 = {};
  #pragma unroll
  for (int kb = 0; kb < 8; ++kb) {
    const int k0 = kb*4 + lh*2;
    v2f a; a.x = hlds[wv][l15*33 + k0]; a.y = hlds[wv][l15*33 + k0 + 1];
    v2f b0, b1;                                  // B[k=c][n=o] = Wt1[o*32+c]
    b0.x = Wt1[l15*CC + k0];        b0.y = Wt1[l15*CC + k0 + 1];
    b1.x = Wt1[(16+l15)*CC + k0];   b1.y = Wt1[(16+l15)*CC + k0 + 1];
    t0 = wmma4(a, b0, t0);
    t1 = wmma4(a, b1, t1);
  }

  // ---- epilogue: relu(tanh(T + bt1 + H)) dot Wout, lane-reduce, store y ----
  const float w0 = Wout[l15], w1 = Wout[16 + l15];
  const float c0 = bt1[l15],  c1 = bt1[16 + l15];
  float contrib[8];
  #pragma unroll
  for (int vv = 0; vv < 8; ++vv) {
    int row = vv + lh*8;
    float a0 = tanhf(t0[vv] + c0 + hlds[wv][row*33 + l15]);
    float a1 = tanhf(t1[vv] + c1 + hlds[wv][row*33 + 16 + l15]);
    contrib[vv] = w0 * fmaxf(a0, 0.f) + w1 * fmaxf(a1, 0.f);
  }
  #pragma unroll
  for (int off = 1; off < 16; off <<= 1) {       // reduce across 16 N-lanes
    #pragma unroll
    for (int vv = 0; vv < 8; ++vv) contrib[vv] += __shfl_xor(contrib[vv], off, 32);
  }
  if (l15 == 0) {
    const float bo = bout[0];
    #pragma unroll
    for (int vv = 0; vv < 8; ++vv) y[tbase + lh*8 + vv] = contrib[vv] + bo;
  }
}

extern "C" void kernel_launch(void* const* d_in, const int* in_sizes, int n_in,
                              void* d_out, int out_size, void* d_ws, size_t ws_size,
                              hipStream_t stream) {
  (void)in_sizes; (void)n_in; (void)out_size; (void)ws_size;
  const float* X      = (const float*)d_in[0];
  const float* adj    = (const float*)d_in[1];
  const float* adjm   = (const float*)d_in[2];
  const float* Theta0 = (const float*)d_in[3];
  const float* bias0  = (const float*)d_in[4];
  const float* Wt0    = (const float*)d_in[5];
  const float* bt0    = (const float*)d_in[6];
  const float* Theta1 = (const float*)d_in[7];
  const float* bias1  = (const float*)d_in[8];
  const float* Wt1    = (const float*)d_in[9];
  const float* bt1    = (const float*)d_in[10];
  const float* Wout   = (const float*)d_in[11];
  const float* bout   = (const float*)d_in[12];
  float* y = (float*)d_out;

  size_t off = 0;
  auto take = [&](size_t bytes) -> void* {
    void* p = (char*)d_ws + off;
    off += (bytes + 255) & ~(size_t)255;
    return p;
  };
  int*   am_ti = (int*)  take((size_t)NN*KK*sizeof(int));
  float* am_tv = (float*)take((size_t)NN*KK*sizeof(float));
  float* am_dw = (float*)take((size_t)NN*sizeof(float));
  int*   a_ti  = (int*)  take((size_t)NN*KK*sizeof(int));
  float* a_tv  = (float*)take((size_t)NN*KK*sizeof(float));
  float* a_dw  = (float*)take((size_t)NN*sizeof(float));
  float* a_s   = (float*)take((size_t)NN*sizeof(float));
  float* h1    = (float*)take((size_t)PP*CC*sizeof(float));   // ~12.3 MB

  topk_kernel<<<NN/8, 256, 0, stream>>>(adjm, am_ti, am_tv, am_dw, nullptr);
  topk_kernel<<<NN/8, 256, 0, stream>>>(adj,  a_ti,  a_tv,  a_dw,  a_s);
  layer0_kernel<<<PP/256, 256, 0, stream>>>(X, am_ti, am_tv, am_dw,
                                            Theta0, bias0, Wt0, bt0, h1);
  layer1_kernel<<<PP/16/WPB, 128, 0, stream>>>(h1, a_ti, a_tv, a_dw, a_s,
                                               Theta1, bias1, Wt1, bt1,
                                               Wout, bout, y);
}